// GraphTransformer_83614423318800
// MI455X (gfx1250) — compile-verified
//
#include <hip/hip_runtime.h>
#include <math.h>
#include <stdint.h>

typedef __attribute__((ext_vector_type(2))) float v2f;
typedef __attribute__((ext_vector_type(4))) float v4f;
typedef __attribute__((ext_vector_type(8))) float v8f;
typedef __attribute__((ext_vector_type(4))) unsigned int v4u;
typedef __attribute__((ext_vector_type(4))) int v4i;
typedef __attribute__((ext_vector_type(8))) int v8i;

#define WMMA_F32(a, b, c) \
  __builtin_amdgcn_wmma_f32_16x16x4_f32(false, (a), false, (b), (short)0, (c), false, false)

// Compiler-only fence (ordering for escaped memory; emits no instructions).
#define OPAQUE_MEM() asm volatile("" ::: "memory")

// Tensor Data Mover builtin exists on both toolchains but with different arity:
// ROCm 7.2 (clang-22): 5 args; amdgpu-toolchain (clang-23): 6 args (extra v8i).
#if __has_builtin(__builtin_amdgcn_tensor_load_to_lds)
#define USE_TDM 1
#else
#define USE_TDM 0
#endif

constexpr int B_ = 2, N_ = 256, IN_ = 256, E_ = 16, H_ = 8, D_ = 64, INNER_ = 512;

// ---------------------------------------------------------------------------
// Kernel 1: y = nodes @ W + b, scattered into heads layout [bh][i][d].
// One wave per 16x16 output tile, fp32 WMMA over K=256 in steps of 4.
// ---------------------------------------------------------------------------
__global__ __launch_bounds__(32) void k_qkv(const float* __restrict__ nodes,
                                            const float* __restrict__ W,
                                            const float* __restrict__ bias,
                                            float* __restrict__ out_heads) {
  const int lane = threadIdx.x;
  const int row0 = blockIdx.x * 16;  // rows = b*N + i, 0..511
  const int col0 = blockIdx.y * 16;  // 0..511
  const int mrow = lane & 15;
  const int kk = (lane >> 4) * 2;
  v8f acc = {};
#pragma unroll 8
  for (int k0 = 0; k0 < IN_; k0 += 4) {
    v2f a = *(const v2f*)(nodes + (row0 + mrow) * IN_ + k0 + kk);
    v2f b;
    b.x = W[(k0 + kk) * INNER_ + col0 + mrow];
    b.y = W[(k0 + kk + 1) * INNER_ + col0 + mrow];
    acc = WMMA_F32(a, b, acc);
  }
  const int col = col0 + mrow;
  const float bs = bias[col];
  const int h = col >> 6, d = col & 63;
#pragma unroll
  for (int r = 0; r < 8; ++r) {
    int row = row0 + r + 8 * (lane >> 4);
    int bb = row >> 8, i = row & 255;
    out_heads[(((bb * H_ + h) * N_ + i) * D_) + d] = acc[r] + bs;
  }
}

// ---------------------------------------------------------------------------
// Kernel 2: qW[bh][i][c] = sum_d q[bh][i][d] * We[c][h*64+d]   (K=64 WMMA)
//           qbe[bh][i]   = sum_d q[bh][i][d] * be[h*64+d]
// ---------------------------------------------------------------------------
__global__ __launch_bounds__(32) void k_qw(const float* __restrict__ q,
                                           const float* __restrict__ We,
                                           const float* __restrict__ be,
                                           float* __restrict__ qW,
                                           float* __restrict__ qbe) {
  const int lane = threadIdx.x;
  const int i0 = blockIdx.x * 16;
  const int bh = blockIdx.y;
  const int h = bh & 7;
  const int mrow = lane & 15;
  const int kk = (lane >> 4) * 2;
  v8f acc = {};
#pragma unroll
  for (int k0 = 0; k0 < D_; k0 += 4) {
    v2f a = *(const v2f*)(q + ((bh * N_ + i0 + mrow) * D_) + k0 + kk);
    v2f b = *(const v2f*)(We + mrow * INNER_ + h * D_ + k0 + kk);
    acc = WMMA_F32(a, b, acc);
  }
#pragma unroll
  for (int r = 0; r < 8; ++r)
    qW[(bh * N_ + i0 + r + 8 * (lane >> 4)) * E_ + mrow] = acc[r];
  if (lane < 16) {
    float s = 0.f;
    const float* qrow = q + (bh * N_ + i0 + lane) * D_;
    const float* berow = be + h * D_;
    for (int d = 0; d < D_; ++d) s += qrow[d] * berow[d];
    qbe[bh * N_ + i0 + lane] = s;
  }
}

// ---------------------------------------------------------------------------
// Kernel 3: sim[bh][i][j] = ( q@k^T + edges . qW + qbe ) / sqrt(D)
// Block = (i-tile, j-tile, b); 8 waves = 8 heads share one LDS edge tile.
// Wave 0 issues a TDM DMA of the 16KB edge tile, overlapped with the q@k^T
// WMMA K-loop; s_wait_tensorcnt + barrier before the edge dot-products.
// The LDS pointer is laundered through inline asm so the compiler cannot
// prove the TDM-written tile is "unwritten" and fold its reads to undef.
// ---------------------------------------------------------------------------
__global__ __launch_bounds__(256) void k_sim(const float* __restrict__ q,
                                             const float* __restrict__ kmat,
                                             const float* __restrict__ edges,
                                             const float* __restrict__ qW,
                                             const float* __restrict__ qbe,
                                             float* __restrict__ sim) {
  __shared__ float eTile[16 * 16 * 16];  // [il][jl][c]  (16 KB, sole LDS object)
  const int i0 = blockIdx.x * 16, j0 = blockIdx.y * 16, b = blockIdx.z;
  const int wave = threadIdx.x >> 5;
  const int lane = threadIdx.x & 31;

  // Launder the LDS base pointer: escapes eTile so the TDM's out-of-band
  // writes cannot be reasoned away (loads below must actually happen).
  const float* ebase_lds = eTile;
  asm volatile("" : "+v"(ebase_lds));

#if USE_TDM
  if (wave == 0) {
    // D# for a 3-D tile: dim0=c (16 contiguous f32), dim1=j (stride 16),
    // dim2=i (stride 256*16=4096). global_addr = tile start; LDS offset 0.
    unsigned long long ga =
        (unsigned long long)(uintptr_t)(edges + (((b * N_ + i0) * N_) + j0) * E_);
    v4u g0;
    g0.x = 1u;                                   // count=1, user mode
    g0.y = 0u;                                   // lds_addr = 0 (eTile base)
    g0.z = (unsigned)(ga & 0xFFFFFFFFu);         // global_addr[31:0]
    g0.w = (unsigned)((ga >> 32) & 0x1FFFFFFu) | (2u << 30);  // [56:32] | type=2
    v8i g1;
    g1[0] = (int)(2u << 16);           // workgroup_mask=0, data_size=4B
    g1[1] = (int)(16u << 16);          // tensor_dim0 = 16   (bits 79:48)
    g1[2] = (int)(16u << 16);          // tensor_dim1 = 16   (bits 111:80)
    g1[3] = (int)(16u << 16);          // tile_dim0  = 16    (bits 127:112)
    g1[4] = (int)((16u << 16) | 16u);  // tile_dim1=16, tile_dim2=16
    g1[5] = 16;                        // tensor_dim0_stride = 16
    g1[6] = (int)(4096u << 16);        // tensor_dim1_stride = 4096 (255:208)
    g1[7] = 0;
    v4i g2;
    g2[0] = 16;  // tensor_dim2 = 16
    g2[1] = 1;   // tensor_dim3 = 1 (unused)
    g2[2] = 0;   // tensor_dim2_stride lo (unused for 3-D)
    g2[3] = 0;   // tile_dim3 = 0
    v4i g3 = {0, 0, 0, 0};
#if defined(__clang_major__) && (__clang_major__ >= 23)
    v8i g4 = {0, 0, 0, 0, 0, 0, 0, 0};  // probe-verified 6-arg form
    __builtin_amdgcn_tensor_load_to_lds(g0, g1, g2, g3, g4, 0);
#else
    __builtin_amdgcn_tensor_load_to_lds(g0, g1, g2, g3, 0);  // ROCm 7.2 form
#endif
  }
  OPAQUE_MEM();  // TDM writes LDS behind the compiler's back
#else
  {
    int t = threadIdx.x;  // 0..255 -> one (il, jl) pair each
    int il = t >> 4, jl = t & 15;
    const float* src = edges + (((b * N_ + i0 + il) * N_) + j0 + jl) * E_;
    float* dst = eTile + (il * 16 + jl) * E_;
#pragma unroll
    for (int c = 0; c < E_; c += 4) *(v4f*)(dst + c) = *(const v4f*)(src + c);
  }
#endif

  // ---- q@k^T WMMA while the edge tile DMA is in flight ----
  const int h = wave;  // wave id == head
  const int bh = b * H_ + h;
  const int mrow = lane & 15;
  const int kk = (lane >> 4) * 2;
  v8f acc = {};
#pragma unroll
  for (int k0 = 0; k0 < D_; k0 += 4) {
    v2f a = *(const v2f*)(q + ((bh * N_ + i0 + mrow) * D_) + k0 + kk);
    v2f bb = *(const v2f*)(kmat + ((bh * N_ + j0 + mrow) * D_) + k0 + kk);
    acc = WMMA_F32(a, bb, acc);
  }

#if USE_TDM
#if __has_builtin(__builtin_amdgcn_s_wait_tensorcnt)
  __builtin_amdgcn_s_wait_tensorcnt(0);  // no-op for waves with TENSORcnt==0
#endif
  OPAQUE_MEM();
#endif
  __syncthreads();
  OPAQUE_MEM();  // eTile contents are now valid; loads below are real

  const int jl = lane & 15;
#pragma unroll
  for (int r = 0; r < 8; ++r) {
    int il = r + 8 * (lane >> 4);
    int gi = i0 + il;
    const float* qWrow = qW + (bh * N_ + gi) * E_;
    const float* erow = ebase_lds + (il * 16 + jl) * E_;  // laundered pointer
    v4f qa = *(const v4f*)(qWrow + 0), qb = *(const v4f*)(qWrow + 4);
    v4f qc = *(const v4f*)(qWrow + 8), qd = *(const v4f*)(qWrow + 12);
    v4f ea = *(const v4f*)(erow + 0), eb = *(const v4f*)(erow + 4);
    v4f ec = *(const v4f*)(erow + 8), ed = *(const v4f*)(erow + 12);
    v4f t = ea * qa + eb * qb + ec * qc + ed * qd;
    float s = qbe[bh * N_ + gi] + t.x + t.y + t.z + t.w;
    sim[(bh * N_ + gi) * N_ + j0 + jl] = (acc[r] + s) * 0.125f;  // 1/sqrt(64)
  }
}

// ---------------------------------------------------------------------------
// Kernel 4: means[i] = mean_j sim[bh=0][i][j]   (faithful to reference's [0])
// ---------------------------------------------------------------------------
__global__ __launch_bounds__(256) void k_means(const float* __restrict__ sim,
                                               float* __restrict__ means) {
  __shared__ float red[256];
  int i = blockIdx.x, j = threadIdx.x;
  red[j] = sim[i * N_ + j];
  __syncthreads();
  for (int s = 128; s > 0; s >>= 1) {
    if (j < s) red[j] += red[j + s];
    __syncthreads();
  }
  if (j == 0) means[i] = red[0] * (1.0f / N_);
}

// ---------------------------------------------------------------------------
// Kernel 5: masked softmax + out = attn@v + aE@We_h + aSum*be_h
// 4 waves per (bh, 16-row i-tile): softmax rows split across waves, the aE
// reduction uses all 128 lanes, then each wave runs the fused WMMA chain for
// its own 16-wide d-tile (attn@v K=256, then aE@We K=16, same accumulator).
// ---------------------------------------------------------------------------
__global__ __launch_bounds__(128) void k_out(const float* __restrict__ sim,
                                             const float* __restrict__ means,
                                             const float* __restrict__ adj,
                                             const float* __restrict__ edges,
                                             const float* __restrict__ vmat,
                                             const float* __restrict__ We,
                                             const float* __restrict__ be,
                                             float* __restrict__ out) {
  __shared__ float attn[16 * 256];  // 16 KB
  __shared__ float aE[16 * 16];
  __shared__ float aSum[16];
  const int i0 = blockIdx.x * 16;
  const int bh = blockIdx.y;
  const int b = bh >> 3, h = bh & 7;
  const int tid = threadIdx.x;
  const int wave = tid >> 5, lane = tid & 31;

  // ---- masked "softmax" with shared bh=0 means and denom==0 guard ----
  for (int il = wave; il < 16; il += 4) {
    const int gi = i0 + il;
    const float m = means[gi];
    const float* srow = sim + (bh * N_ + gi) * N_;
    const float* arow = adj + (b * N_ + gi) * N_;
    float rs = 0.f;
    for (int j = lane; j < N_; j += 32) {
      float x = __expf(srow[j] - m) * arow[j];
      attn[il * N_ + j] = x;
      rs += x;
    }
    for (int off = 16; off > 0; off >>= 1) rs += __shfl_xor(rs, off, 32);
    float denom = (rs == 0.f) ? 1.f : rs;
    float inv = 1.f / denom;
    for (int j = lane; j < N_; j += 32) attn[il * N_ + j] *= inv;
    if (lane == 0) aSum[il] = rs * inv;
  }
  __syncthreads();

  // ---- aE[i][c] = sum_j attn[i][j] * edges[b][i][j][c], 2 c's per lane ----
  {
    const int il = tid & 15, cg = tid >> 4;  // cg = 0..7 -> c pair cg*2
    v2f acc2 = {};
    const float* ebase = edges + ((b * N_ + i0 + il) * N_) * E_ + cg * 2;
    for (int j = 0; j < N_; ++j) {
      float a = attn[il * N_ + j];
      int jp = (j + 16 < N_) ? (j + 16) : (N_ - 1);  // branch-free prefetch idx
      __builtin_prefetch(ebase + jp * E_, 0, 0);
      v2f e = *(const v2f*)(ebase + j * E_);
      acc2 += a * e;
    }
    aE[il * 16 + cg * 2] = acc2.x;
    aE[il * 16 + cg * 2 + 1] = acc2.y;
  }
  __syncthreads();

  // ---- out tile (one d-tile per wave): WMMA over v (K=256) + aE@We (K=16) --
  const int mrow = lane & 15;
  const int kk = (lane >> 4) * 2;
  const int d0 = wave * 16;
  v8f acc = {};
#pragma unroll 8
  for (int k0 = 0; k0 < N_; k0 += 4) {
    v2f a = *(const v2f*)(attn + mrow * N_ + k0 + kk);
    v2f bb;
    bb.x = vmat[(bh * N_ + k0 + kk) * D_ + d0 + mrow];
    bb.y = vmat[(bh * N_ + k0 + kk + 1) * D_ + d0 + mrow];
    acc = WMMA_F32(a, bb, acc);
  }
#pragma unroll
  for (int k0 = 0; k0 < E_; k0 += 4) {
    v2f a = *(const v2f*)(aE + mrow * 16 + k0 + kk);
    v2f bb;
    bb.x = We[(k0 + kk) * INNER_ + h * D_ + d0 + mrow];
    bb.y = We[(k0 + kk + 1) * INNER_ + h * D_ + d0 + mrow];
    acc = WMMA_F32(a, bb, acc);
  }
  const int dcol = h * D_ + d0 + mrow;
  const float bs = be[dcol];
#pragma unroll
  for (int r = 0; r < 8; ++r) {
    int il = r + 8 * (lane >> 4);
    out[((b * N_ + i0 + il) * INNER_) + dcol] = acc[r] + aSum[il] * bs;
  }
}

// ---------------------------------------------------------------------------
extern "C" void kernel_launch(void* const* d_in, const int* in_sizes, int n_in,
                              void* d_out, int out_size, void* d_ws, size_t ws_size,
                              hipStream_t stream) {
  const float* nodes = (const float*)d_in[0];
  const float* edges = (const float*)d_in[1];
  const float* adj   = (const float*)d_in[2];
  const float* Wq = (const float*)d_in[3];
  const float* bq = (const float*)d_in[4];
  const float* Wk = (const float*)d_in[5];
  const float* bk = (const float*)d_in[6];
  const float* Wv = (const float*)d_in[7];
  const float* bv = (const float*)d_in[8];
  const float* We = (const float*)d_in[9];
  const float* be = (const float*)d_in[10];
  float* out = (float*)d_out;

  float* ws = (float*)d_ws;                 // 7.6 MB total, L2-resident
  float* qf = ws;                           // [bh][i][d]  1 MB
  float* kf = qf + 16 * 256 * 64;           // 1 MB
  float* vf = kf + 16 * 256 * 64;           // 1 MB
  float* qW = vf + 16 * 256 * 64;           // [bh][i][c]  256 KB
  float* simf = qW + 16 * 256 * 16;         // [bh][i][j]  4 MB
  float* meanf = simf + 16 * 256 * 256;     // [i]         1 KB
  float* qbe = meanf + 256;                 // [bh][i]     16 KB

  dim3 b32(32), b128(128), b256(256);

  // QKV projections (fp32 WMMA GEMMs)
  k_qkv<<<dim3(32, 32), b32, 0, stream>>>(nodes, Wq, bq, qf);
  k_qkv<<<dim3(32, 32), b32, 0, stream>>>(nodes, Wk, bk, kf);
  k_qkv<<<dim3(32, 32), b32, 0, stream>>>(nodes, Wv, bv, vf);

  // qW = q . We_h^T  and  qbe = q . be_h
  k_qw<<<dim3(16, 16), b32, 0, stream>>>(qf, We, be, qW, qbe);

  // sim = (q@k^T + edges.qW + qbe) / sqrt(D), edge tile via TDM into LDS
  k_sim<<<dim3(16, 16, 2), b256, 0, stream>>>(qf, kf, edges, qW, qbe, simf);

  // row means of sim at bh=0 (reference's [0] slice)
  k_means<<<dim3(256), b256, 0, stream>>>(simf, meanf);

  // masked softmax + fused output GEMMs
  k_out<<<dim3(16, 16), b128, 0, stream>>>(simf, meanf, adj, edges, vf, We, be, out);
}